// PatchQuantumGenerator_4578435138168
// MI455X (gfx1250) — compile-verified
//
#include <hip/hip_runtime.h>
#include <math.h>

typedef __attribute__((ext_vector_type(2))) float v2f;
typedef __attribute__((ext_vector_type(8))) float v8f;

#define N_QUBITS 5
#define Q_DEPTH  6
#define N_GEN    4
#define DIM      32
#define PATCH    16

// ---------------------------------------------------------------------------
// Kernel 1: build the fixed 32x64 weight matrix (16 output rows of each U_g),
// stored directly in V_WMMA_F32_16X16X4_F32 B-fragment lane order:
//   wfrag[((g*8 + ks)*32 + lane)*2 + v]  =  U_g[n = lane&15, k = ks*4 + v + 2*(lane>>4)]
// One thread per (generator g, basis column i): simulate the circuit on e_i.
// ---------------------------------------------------------------------------
__global__ __launch_bounds__(128) void qgen_build_weights(
    const float* __restrict__ qp,   // (N_GEN, Q_DEPTH*N_QUBITS)
    float* __restrict__ wfrag)      // 4*8*32*2 = 2048 floats
{
    const int t = threadIdx.x;          // 0..127
    const int g = t >> 5;
    const int i = t & 31;               // basis column

    float st[DIM];
#pragma unroll
    for (int j = 0; j < DIM; ++j) st[j] = (j == i) ? 1.0f : 0.0f;

#pragma unroll
    for (int d = 0; d < Q_DEPTH; ++d) {
#pragma unroll
        for (int q = 0; q < N_QUBITS; ++q) {
            const float th = 0.5f * qp[g * (Q_DEPTH * N_QUBITS) + d * N_QUBITS + q];
            const float c = cosf(th);
            const float s = sinf(th);
            const int bit = 1 << (4 - q);   // qubit q is bit (4-q) of the index
#pragma unroll
            for (int j0 = 0; j0 < DIM; ++j0) {
                if (j0 & bit) continue;
                const int j1 = j0 | bit;
                const float a0 = st[j0], a1 = st[j1];
                st[j0] = c * a0 - s * a1;
                st[j1] = s * a0 + c * a1;
            }
        }
        // CZ chain sign: prod_q (1 - 2*b_q*b_{q+1})
#pragma unroll
        for (int j = 0; j < DIM; ++j) {
            float sg = 1.0f;
#pragma unroll
            for (int q = 0; q < N_QUBITS - 1; ++q) {
                const int b0 = (j >> (4 - q)) & 1;
                const int b1 = (j >> (3 - q)) & 1;
                if (b0 & b1) sg = -sg;
            }
            st[j] *= sg;
        }
    }

    // Thread holds column i of U_g. Scatter rows n=0..15 into B-fragment order.
    const int ks   = i >> 2;            // k-slice (K step of 4)
    const int v    = i & 1;             // which of the 2 B VGPRs
    const int half = (i >> 1) & 1;      // lanes 0-15 vs 16-31
#pragma unroll
    for (int n = 0; n < PATCH; ++n) {
        const int lane = half * 16 + n;
        wfrag[((g * 8 + ks) * 32 + lane) * 2 + v] = st[n];
    }
}

// ---------------------------------------------------------------------------
// Kernel 2: main batched GEMM + normalize.
//   out[b, g*16+n] = p / max_n(p),  p = (U_g s(b))_n^2
// Each wave handles 16 batch rows; 4 generators x 8 K-slices of
// v_wmma_f32_16x16x4_f32. 256 threads = 8 waves = 128 rows per block.
// ---------------------------------------------------------------------------
__device__ __forceinline__ float amp5(int k, const float c[5], const float s[5]) {
    float a = 1.0f;
#pragma unroll
    for (int q = 0; q < N_QUBITS; ++q)
        a *= ((k >> (4 - q)) & 1) ? s[q] : c[q];
    return a;
}

__global__ __launch_bounds__(256) void qgen_main(
    const float* __restrict__ x,      // (B, 5)
    const float* __restrict__ wfrag,  // 2048 floats, B-fragment order
    float* __restrict__ out)          // (B, 64)
{
    const int lane = threadIdx.x & 31;
    const int wv   = threadIdx.x >> 5;
    const int b0   = blockIdx.x * 128 + wv * 16;   // first batch row of this wave's tile
    const int row  = b0 + (lane & 15);             // this lane's A-matrix row (M)

    // Per-row rotation factors
    float c[5], s[5];
#pragma unroll
    for (int q = 0; q < N_QUBITS; ++q) {
        const float h = 0.5f * x[(size_t)row * N_QUBITS + q];
        c[q] = cosf(h);
        s[q] = sinf(h);
    }

    // A fragments for all 8 K-slices: lane holds S[row][ks*4 + 2*(lane>>4) + {0,1}]
    const int half2 = (lane >> 4) << 1;   // 0 or 2
    v2f A[8];
#pragma unroll
    for (int ks = 0; ks < 8; ++ks) {
        const int k0 = ks * 4 + half2;
        A[ks].x = amp5(k0,     c, s);
        A[ks].y = amp5(k0 + 1, c, s);
    }

    const int halfsel = (lane >> 4) << 3; // +8 row offset for upper half-lanes

#pragma unroll
    for (int g = 0; g < N_GEN; ++g) {
        v8f acc = {};
#pragma unroll
        for (int ks = 0; ks < 8; ++ks) {
            const float* wp = wfrag + ((size_t)(g * 8 + ks) * 32 + lane) * 2;
            v2f Bf;
            Bf.x = wp[0];
            Bf.y = wp[1];
            acc = __builtin_amdgcn_wmma_f32_16x16x4_f32(
                /*neg_a=*/false, A[ks], /*neg_b=*/false, Bf,
                /*c_mod=*/(short)0, acc, /*reuse_a=*/false, /*reuse_b=*/false);
        }
        // acc VGPR v: lanes 0-15 -> (M=v, N=lane), lanes 16-31 -> (M=v+8, N=lane-16).
        // out = p / max(p) over the 16 N values (sum normalization cancels).
#pragma unroll
        for (int v = 0; v < 8; ++v) {
            const float d = acc[v];
            const float p = d * d;
            float m = p;
            m = fmaxf(m, __shfl_xor(m, 1, 32));
            m = fmaxf(m, __shfl_xor(m, 2, 32));
            m = fmaxf(m, __shfl_xor(m, 4, 32));
            m = fmaxf(m, __shfl_xor(m, 8, 32));
            const int r = v + halfsel;
            out[(size_t)(b0 + r) * (N_GEN * PATCH) + g * PATCH + (lane & 15)] = p / m;
        }
    }
}

extern "C" void kernel_launch(void* const* d_in, const int* in_sizes, int n_in,
                              void* d_out, int out_size, void* d_ws, size_t ws_size,
                              hipStream_t stream) {
    const float* x  = (const float*)d_in[0];   // (262144, 5) fp32
    const float* qp = (const float*)d_in[1];   // (4, 30) fp32
    float* out   = (float*)d_out;              // (262144, 64) fp32
    float* wfrag = (float*)d_ws;               // 2048 floats = 8 KB scratch

    const int B = in_sizes[0] / N_QUBITS;      // 262144

    qgen_build_weights<<<1, 128, 0, stream>>>(qp, wfrag);
    qgen_main<<<B / 128, 256, 0, stream>>>(x, wfrag, out);
}